// TrueMambaS6Block_6914897346747
// MI455X (gfx1250) — compile-verified
//
#include <hip/hip_runtime.h>
#include <math.h>

// Mamba S6 selective scan for MI455X (gfx1250, wave32).
// Pipeline: [WMMA proj: delta/B/C] -> [parallel-state sequential scan] -> [WMMA out-proj in-place].

typedef __attribute__((ext_vector_type(2))) float v2f;
typedef __attribute__((ext_vector_type(8))) float v8f;

#define BSZ   16
#define SEQL  4096
#define DDIM  64
#define NST   16
#define MROWS (BSZ * SEQL)   // 65536 token rows

// D = A(16x4,f32) * B(4x16,f32) + C(16x16,f32)
__device__ __forceinline__ v8f wmma_k4(v2f a, v2f b, v8f c) {
  return __builtin_amdgcn_wmma_f32_16x16x4_f32(
      /*neg_a=*/false, a, /*neg_b=*/false, b,
      /*c_mod=*/(short)0, c, /*reuse_a=*/false, /*reuse_b=*/false);
}

// Branch-free, numerically stable softplus using only the native
// transcendental pipe (v_exp_f32 + v_log_f32), no frexp/log1p libm path:
//   softplus(z) = max(z,0) + log(1 + exp(-|z|))
__device__ __forceinline__ float softplus_f(float z) {
  return fmaxf(z, 0.0f) + __logf(1.0f + __expf(-fabsf(z)));
}

// ---------------------------------------------------------------------------
// Kernel 1: fused input projections.
//   delta[m, 0:64] = softplus(x[m,:] . Wd[e,:] + bd[e])
//   Bt[m, 0:16]    = x[m,:] . WB[n,:]
//   Ct[m, 0:16]    = x[m,:] . WC[n,:]
// One wave handles one 16-row tile; A fragment (16 K-chunks) reused across
// all 6 sixteen-column output tiles. 96 v_wmma per wave.
// ---------------------------------------------------------------------------
__global__ __launch_bounds__(256) void mamba_proj_kernel(
    const float* __restrict__ x,
    const float* __restrict__ Wd, const float* __restrict__ bd,
    const float* __restrict__ WB, const float* __restrict__ WC,
    float* __restrict__ dlt, float* __restrict__ Bt, float* __restrict__ Ct)
{
  const int lane = threadIdx.x & 31;
  const int wid  = threadIdx.x >> 5;
  const int tile = blockIdx.x * 8 + wid;       // 16-row tile index
  const int row0 = tile << 4;
  const int m    = lane & 15;                  // row within tile (A frag)
  const int n    = lane & 15;                  // col within tile (B frag / store)
  const int hi   = lane >> 4;                  // selects K pair {0,1} vs {2,3}

  // A fragment: X tile, 16 chunks of K=4
  const float* xrow = x + (size_t)(row0 + m) * DDIM + hi * 2;
  v2f a[16];
#pragma unroll
  for (int kk = 0; kk < 16; ++kk)
    a[kk] = *(const v2f*)(xrow + kk * 4);

  // ---- delta: 4 column tiles of W_delta, bias + softplus on store ----
#pragma unroll
  for (int ct = 0; ct < 4; ++ct) {
    const int col = ct * 16 + n;
    const float* wrow = Wd + (size_t)col * DDIM + hi * 2;  // B[k][n] = Wd[col][k]
    v8f acc = {};
#pragma unroll
    for (int kk = 0; kk < 16; ++kk)
      acc = wmma_k4(a[kk], *(const v2f*)(wrow + kk * 4), acc);
    const float bias = bd[col];
#pragma unroll
    for (int r = 0; r < 8; ++r) {
      const int gm = row0 + r + 8 * hi;        // C/D layout: vgpr r -> M = r + 8*hi
      dlt[(size_t)gm * DDIM + col] = softplus_f(acc[r] + bias);
    }
  }

  // ---- Bt: one 16-col tile ----
  {
    const float* wrow = WB + (size_t)n * DDIM + hi * 2;
    v8f acc = {};
#pragma unroll
    for (int kk = 0; kk < 16; ++kk)
      acc = wmma_k4(a[kk], *(const v2f*)(wrow + kk * 4), acc);
#pragma unroll
    for (int r = 0; r < 8; ++r)
      Bt[(size_t)(row0 + r + 8 * hi) * NST + n] = acc[r];
  }

  // ---- Ct: one 16-col tile ----
  {
    const float* wrow = WC + (size_t)n * DDIM + hi * 2;
    v8f acc = {};
#pragma unroll
    for (int kk = 0; kk < 16; ++kk)
      acc = wmma_k4(a[kk], *(const v2f*)(wrow + kk * 4), acc);
#pragma unroll
    for (int r = 0; r < 8; ++r)
      Ct[(size_t)(row0 + r + 8 * hi) * NST + n] = acc[r];
  }
}

// ---------------------------------------------------------------------------
// Kernel 2: sequential selective scan. One wave per (b, d-pair):
//   lanes = (n in 0..15) x (j in 0..1), d = 2*dp + j.
// Loop-carried chain is a single FMA on h; loads for t+1 are issued before
// the t computation (software pipelining), and the C.h reduction over n uses
// 4 xor-shuffles within each 16-lane group.
// ---------------------------------------------------------------------------
__global__ __launch_bounds__(32) void mamba_scan_kernel(
    const float* __restrict__ x, const float* __restrict__ A_log,
    const float* __restrict__ D_skip,
    const float* __restrict__ dlt, const float* __restrict__ Bt,
    const float* __restrict__ Ct, float* __restrict__ y)
{
  const int lane = threadIdx.x;
  const int n  = lane & 15;
  const int j  = lane >> 4;
  const int dp = blockIdx.x & 31;              // D/2 = 32 d-pairs
  const int b  = blockIdx.x >> 5;
  const int d  = dp * 2 + j;

  const float Areal = -__expf(A_log[d * NST + n]);
  const float invA  = 1.0f / Areal;
  const float dsk   = D_skip[d];

  const size_t baseD = (size_t)b * SEQL * DDIM + d;
  const size_t baseN = (size_t)b * SEQL * NST + n;
  const float* dptr = dlt + baseD;
  const float* xptr = x   + baseD;
  const float* bptr = Bt  + baseN;
  const float* cptr = Ct  + baseN;
  float*       yptr = y   + baseD;

  float h  = 0.0f;
  float dt = dptr[0], xv = xptr[0], bv = bptr[0], cv = cptr[0];

  for (int t = 0; t < SEQL; ++t) {
    // prefetch next step (clamped index avoids a branch in the loop body)
    const int tn = (t + 1 < SEQL) ? (t + 1) : t;
    const float dt2 = dptr[(size_t)tn * DDIM];
    const float xv2 = xptr[(size_t)tn * DDIM];
    const float bv2 = bptr[(size_t)tn * NST];
    const float cv2 = cptr[(size_t)tn * NST];

    const float Abar = __expf(dt * Areal);               // off the h-chain
    h = fmaf(Abar, h, (Abar - 1.0f) * invA * bv * xv);   // the only serial dep

    float acc = cv * h;                                  // reduce over n (16 lanes)
    acc += __shfl_xor(acc, 1, 32);
    acc += __shfl_xor(acc, 2, 32);
    acc += __shfl_xor(acc, 4, 32);
    acc += __shfl_xor(acc, 8, 32);
    if (n == 0)
      yptr[(size_t)t * DDIM] = fmaf(dsk, xv, acc);

    dt = dt2; xv = xv2; bv = bv2; cv = cv2;
  }
}

// ---------------------------------------------------------------------------
// Kernel 3: output projection, IN PLACE on d_out (y -> y.Wo^T + bo).
// Safe: each wave fully loads its own 16 rows into registers before storing
// back to those same rows; no other wave touches them.
// ---------------------------------------------------------------------------
__global__ __launch_bounds__(256) void mamba_outproj_kernel(
    float* __restrict__ y, const float* __restrict__ Wo,
    const float* __restrict__ bo)
{
  const int lane = threadIdx.x & 31;
  const int wid  = threadIdx.x >> 5;
  const int tile = blockIdx.x * 8 + wid;
  const int row0 = tile << 4;
  const int m    = lane & 15;
  const int n    = lane & 15;
  const int hi   = lane >> 4;

  const float* yrow = y + (size_t)(row0 + m) * DDIM + hi * 2;
  v2f a[16];
#pragma unroll
  for (int kk = 0; kk < 16; ++kk)
    a[kk] = *(const v2f*)(yrow + kk * 4);

  v8f acc[4];
#pragma unroll
  for (int ct = 0; ct < 4; ++ct) {
    const int col = ct * 16 + n;
    const float* wrow = Wo + (size_t)col * DDIM + hi * 2;
    acc[ct] = (v8f){};
#pragma unroll
    for (int kk = 0; kk < 16; ++kk)
      acc[ct] = wmma_k4(a[kk], *(const v2f*)(wrow + kk * 4), acc[ct]);
  }

#pragma unroll
  for (int ct = 0; ct < 4; ++ct) {
    const int col = ct * 16 + n;
    const float bias = bo[col];
#pragma unroll
    for (int r = 0; r < 8; ++r)
      y[(size_t)(row0 + r + 8 * hi) * DDIM + col] = acc[ct][r] + bias;
  }
}

// ---------------------------------------------------------------------------
extern "C" void kernel_launch(void* const* d_in, const int* in_sizes, int n_in,
                              void* d_out, int out_size, void* d_ws, size_t ws_size,
                              hipStream_t stream) {
  const float* x      = (const float*)d_in[0];  // [B,L,D]
  const float* A_log  = (const float*)d_in[1];  // [D,N]
  const float* D_skip = (const float*)d_in[2];  // [D]
  const float* Wo     = (const float*)d_in[3];  // [D,D]
  const float* bo     = (const float*)d_in[4];  // [D]
  const float* Wd     = (const float*)d_in[5];  // [D,D]
  const float* bd     = (const float*)d_in[6];  // [D]
  const float* WB     = (const float*)d_in[7];  // [N,D]
  const float* WC     = (const float*)d_in[8];  // [N,D]
  float* out = (float*)d_out;                   // [B,L,D]

  // workspace: delta (16MB) + Bt (4MB) + Ct (4MB) = 24MB
  float* dlt = (float*)d_ws;
  float* Btw = dlt + (size_t)MROWS * DDIM;
  float* Ctw = Btw + (size_t)MROWS * NST;

  const int gemm_blocks = MROWS / 16 / 8;       // 4096 tiles / 8 waves = 512

  mamba_proj_kernel<<<gemm_blocks, 256, 0, stream>>>(x, Wd, bd, WB, WC, dlt, Btw, Ctw);
  mamba_scan_kernel<<<BSZ * (DDIM / 2), 32, 0, stream>>>(x, A_log, D_skip, dlt, Btw, Ctw, out);
  mamba_outproj_kernel<<<gemm_blocks, 256, 0, stream>>>(out, Wo, bo);
}